// torch_model_18820546691190
// MI455X (gfx1250) — compile-verified
//
#include <hip/hip_runtime.h>

typedef __attribute__((ext_vector_type(2))) float v2f;
typedef __attribute__((ext_vector_type(8))) float v8f;

#define QDIM 4096
#define NDIM 65536
#define DDIM 128
#define KTOP 21
#define WAVES 8
#define COLS_PER_WAVE 32
#define COLS_PER_WG (WAVES * COLS_PER_WAVE) /* 256 */
#define A_STRIDE 132                        /* padded LDS row stride (floats) */
#define A_BUF (16 * A_STRIDE)
#define NCHUNK (QDIM / 16) /* 256 */

// Attempt one top-k insertion for this lane's column.
// tv/ti point at topv[wave][0][lane] / topi[wave][0][lane], k-stride = 32.
__device__ __forceinline__ void topk_try(float s, int qidx,
                                         float* __restrict__ tv, int* __restrict__ ti,
                                         float& minv, int& minpos) {
    if (s > minv) {
        tv[minpos * COLS_PER_WAVE] = s;
        ti[minpos * COLS_PER_WAVE] = qidx;
        float mv = tv[0];
        int mp = 0;
#pragma unroll 1
        for (int k = 1; k < KTOP; ++k) {
            float v = tv[k * COLS_PER_WAVE];
            if (v < mv) { mv = v; mp = k; }
        }
        minv = mv;
        minpos = mp;
    }
}

__global__ __launch_bounds__(256) void fused_gemm_topk(const float* __restrict__ xq,
                                                       const float* __restrict__ xb,
                                                       int* __restrict__ out) {
    __shared__ float lds_a[2][A_BUF];                    // double-buffered A chunk
    __shared__ float topv[WAVES][KTOP][COLS_PER_WAVE];   // per-column top-K values
    __shared__ int   topi[WAVES][KTOP][COLS_PER_WAVE];   // per-column top-K indices

    const int tid  = threadIdx.x;
    const int wave = tid >> 5;
    const int lane = tid & 31;
    const int ln   = lane & 15;
    const int lh   = lane >> 4;

    const int strip  = blockIdx.x * COLS_PER_WG + wave * COLS_PER_WAVE;
    const int my_col = strip + lane; // column this lane owns for top-k

    // ---- Load both B panels once (2 x 16 columns x 128 K) -> 64 v2f fragments/lane.
    v2f b0[32], b1[32];
    {
        const float* r0 = xb + (size_t)(strip + ln) * DDIM + 2 * lh;
        const float* r1 = xb + (size_t)(strip + 16 + ln) * DDIM + 2 * lh;
#pragma unroll
        for (int j = 0; j < 32; ++j) {
            b0[j] = *(const v2f*)(r0 + 4 * j);
            b1[j] = *(const v2f*)(r1 + 4 * j);
        }
    }

    // ---- Init per-column top-K state (every lane owns one column).
    float minv  = -INFINITY;
    int   minpos = 0;
    float* tv = &topv[wave][0][lane];
    int*   ti = &topi[wave][0][lane];
#pragma unroll
    for (int k = 0; k < KTOP; ++k) {
        tv[k * COLS_PER_WAVE] = -INFINITY;
        ti[k * COLS_PER_WAVE] = 0;
    }

    // ---- A-staging thread mapping: 16 rows x 128 floats, 8 floats/thread.
    const int arow = tid >> 4;  // 0..15
    const int apar = tid & 15;  // 0..15

    // Prologue: stage chunk 0 into buffer 0.
    {
        const float4* src = (const float4*)(xq + (size_t)arow * DDIM + apar * 8);
        float4 s0 = src[0], s1 = src[1];
        float* dst = &lds_a[0][arow * A_STRIDE + apar * 8];
        *(float4*)dst       = s0;
        *(float4*)(dst + 4) = s1;
    }

    for (int i = 0; i < NCHUNK; ++i) {
        const int cur = i & 1;
        const int q0  = i * 16;
        const bool more = (i + 1 < NCHUNK);

        __syncthreads(); // current buffer's stores (prev iteration) now visible;
                         // loadcnt is already 0 here, so the fence drain is cheap.

        // Issue next chunk's global loads AFTER the barrier so their latency is
        // covered by the 64-WMMA block below (the barrier fence cannot drain them).
        float4 s0 = {}, s1 = {};
        if (more) {
            const float4* src = (const float4*)(xq + (size_t)(q0 + 16 + arow) * DDIM + apar * 8);
            s0 = src[0];
            s1 = src[1];
        }

        // ---- 2 tiles x (16x16) = A(16x128) x B(128x32), 64 WMMAs, 4 accumulator chains.
        v8f c0a = {}, c0b = {}, c1a = {}, c1b = {};
        const float* ap = &lds_a[cur][ln * A_STRIDE + 2 * lh];
#pragma unroll
        for (int j = 0; j < 32; j += 2) {
            v2f a0 = *(const v2f*)(ap + 4 * j);
            v2f a1 = *(const v2f*)(ap + 4 * j + 4);
            c0a = __builtin_amdgcn_wmma_f32_16x16x4_f32(false, a0, false, b0[j],
                                                        (short)0, c0a, false, false);
            c1a = __builtin_amdgcn_wmma_f32_16x16x4_f32(false, a0, false, b1[j],
                                                        (short)0, c1a, false, false);
            c0b = __builtin_amdgcn_wmma_f32_16x16x4_f32(false, a1, false, b0[j + 1],
                                                        (short)0, c0b, false, false);
            c1b = __builtin_amdgcn_wmma_f32_16x16x4_f32(false, a1, false, b1[j + 1],
                                                        (short)0, c1b, false, false);
        }

        // Stage next chunk into the other buffer (loadcnt wait lands here, after
        // the WMMA block has covered the global-load latency).
        if (more) {
            float* dst = &lds_a[cur ^ 1][arow * A_STRIDE + apar * 8];
            *(float4*)dst       = s0;
            *(float4*)(dst + 4) = s1;
        }

        // ---- Top-k epilogue, fully in registers + one shfl per row.
        // C layout: lane (ln,lh) holds rows M = r + 8*lh of column ln (tile0 in c0*, tile1 in c1*).
        // Lane L<16 owns tile0 col L; lane L>=16 owns tile1 col L-16.
        // Exchange the non-owned halves with partner lane (L ^ 16).
#pragma unroll
        for (int r = 0; r < 8; ++r) {
            float t0 = c0a[r] + c0b[r]; // tile0, row r+8*lh, col ln
            float t1 = c1a[r] + c1b[r]; // tile1, row r+8*lh, col ln
            float send = lh ? t0 : t1;  // give partner the half it owns
            float recv = __shfl_xor(send, 16, 32);
            float lo = lh ? recv : t0;  // my column, row r
            float hi = lh ? t1 : recv;  // my column, row r+8
            topk_try(lo, q0 + r,     tv, ti, minv, minpos);
            topk_try(hi, q0 + 8 + r, tv, ti, minv, minpos);
        }
    }

    // ---- Sort descending (selection sort on 21 entries) and emit indices.
#pragma unroll 1
    for (int k = 0; k < KTOP; ++k) {
        float best = tv[k * COLS_PER_WAVE];
        int   bp   = k;
#pragma unroll 1
        for (int j = k + 1; j < KTOP; ++j) {
            float v = tv[j * COLS_PER_WAVE];
            if (v > best) { best = v; bp = j; }
        }
        float tmpv = tv[k * COLS_PER_WAVE];
        int   tmpi = ti[k * COLS_PER_WAVE];
        tv[k * COLS_PER_WAVE]  = tv[bp * COLS_PER_WAVE];
        ti[k * COLS_PER_WAVE]  = ti[bp * COLS_PER_WAVE];
        tv[bp * COLS_PER_WAVE] = tmpv;
        ti[bp * COLS_PER_WAVE] = tmpi;
        out[(size_t)k * NDIM + my_col] = ti[k * COLS_PER_WAVE];
    }
}

extern "C" void kernel_launch(void* const* d_in, const int* in_sizes, int n_in,
                              void* d_out, int out_size, void* d_ws, size_t ws_size,
                              hipStream_t stream) {
    (void)in_sizes; (void)n_in; (void)out_size; (void)d_ws; (void)ws_size;
    const float* xq = (const float*)d_in[0];
    const float* xb = (const float*)d_in[1];
    int* out = (int*)d_out;

    dim3 grid(NDIM / COLS_PER_WG); // 256 workgroups
    dim3 block(256);               // 8 wave32
    fused_gemm_topk<<<grid, block, 0, stream>>>(xq, xb, out);
}